// GraphBlock_39926015983819
// MI455X (gfx1250) — compile-verified
//
#include <hip/hip_runtime.h>
#include <hip/hip_bf16.h>

typedef __attribute__((ext_vector_type(2))) float v2f;
typedef __attribute__((ext_vector_type(8))) float v8f;

#define IN_DIM       256
#define OUT_DIM      128
#define N_PAIRS      (IN_DIM / 2)      // 128 K-pairs
#define LDS_P_STRIDE 144               // float2 units; 288 dwords = 32 mod 64 banks
                                       // -> half-waves (pair p vs p+1) use disjoint banks

static __device__ __forceinline__ v8f wmma_f32(v2f a, v2f b, v8f c) {
    return __builtin_amdgcn_wmma_f32_16x16x4_f32(
        false, a, false, b, (short)0, c, false, false);
}

// One wave computes a 32-row x 64-col output block:
// 2 M-tiles (rows mBase..mBase+31) x 4 N-tiles (cols cBase..cBase+63).
// acc = 2*4*8 = 64 VGPRs; each B fragment (one 64-bit DS load) feeds 2 WMMAs.
template <bool GUARD>
static __device__ __forceinline__ void compute_block(
    const float* __restrict__ X, const float2* __restrict__ ldsP,
    float* __restrict__ H, int mBase, int cBase, int nRows, int lane) {

    const int lane16 = lane & 15;
    const int hi     = lane >> 4;          // 0: K+0/K+1 half, 1: K+2/K+3 half
    const int koff   = hi << 1;

    int r0 = mBase + lane16;
    int r1 = mBase + 16 + lane16;
    if (GUARD) {
        r0 = (r0 < nRows) ? r0 : (nRows - 1);
        r1 = (r1 < nRows) ? r1 : (nRows - 1);
    }
    const float* a0p = X + (size_t)r0 * IN_DIM + koff;
    const float* a1p = X + (size_t)r1 * IN_DIM + koff;

    v8f acc0[4] = {};
    v8f acc1[4] = {};

    float2 c0 = *(const float2*)(a0p);
    float2 c1 = *(const float2*)(a1p);

    #pragma unroll 2
    for (int k = 0; k < IN_DIM - 4; k += 4) {
        // prefetch next A fragments while this step computes
        const float2 nx0 = *(const float2*)(a0p + k + 4);
        const float2 nx1 = *(const float2*)(a1p + k + 4);

        v2f A0; A0.x = c0.x; A0.y = c0.y;
        v2f A1; A1.x = c1.x; A1.y = c1.y;
        const float2* bBase =
            ldsP + ((k >> 1) + hi) * LDS_P_STRIDE + cBase + lane16;
        #pragma unroll
        for (int n = 0; n < 4; ++n) {
            const v2f b = *(const v2f*)(bBase + n * 16);  // one ds_load_b64
            acc0[n] = wmma_f32(A0, b, acc0[n]);
            acc1[n] = wmma_f32(A1, b, acc1[n]);
        }
        c0 = nx0; c1 = nx1;
    }
    {   // peeled final K-step (no prefetch -> no OOB read)
        const int k = IN_DIM - 4;
        v2f A0; A0.x = c0.x; A0.y = c0.y;
        v2f A1; A1.x = c1.x; A1.y = c1.y;
        const float2* bBase =
            ldsP + ((k >> 1) + hi) * LDS_P_STRIDE + cBase + lane16;
        #pragma unroll
        for (int n = 0; n < 4; ++n) {
            const v2f b = *(const v2f*)(bBase + n * 16);
            acc0[n] = wmma_f32(A0, b, acc0[n]);
            acc1[n] = wmma_f32(A1, b, acc1[n]);
        }
    }

    // C/D layout: VGPR g -> row base+g (lanes 0-15) / base+g+8 (lanes 16-31)
    const int rowOff = hi << 3;
    #pragma unroll
    for (int n = 0; n < 4; ++n) {
        const int col = cBase + n * 16 + lane16;
        #pragma unroll
        for (int g = 0; g < 8; ++g) {
            const int row0 = mBase + rowOff + g;
            if (!GUARD || row0 < nRows)
                H[(size_t)row0 * OUT_DIM + col] = acc0[n][g];
        }
        #pragma unroll
        for (int g = 0; g < 8; ++g) {
            const int row1 = mBase + 16 + rowOff + g;
            if (!GUARD || row1 < nRows)
                H[(size_t)row1 * OUT_DIM + col] = acc1[n][g];
        }
    }
}

// -------------------------------------------------------------------------
// Kernel 1: h = X @ W via V_WMMA_F32_16X16X4_F32.
// W staged pair-interleaved in LDS: ldsP[p][c] = {W[2p][c], W[2p+1][c]}.
// Block (8 waves) covers a 128-row x 128-col super-tile:
//   waveN = wave&1 -> column half, waveM = wave>>1 -> 32-row sub-block.
// -------------------------------------------------------------------------
__global__ void gemm_wmma_f32(const float* __restrict__ X,
                              const float* __restrict__ W,
                              float* __restrict__ H, int nRows) {
    extern __shared__ float2 ldsP[];   // [N_PAIRS][LDS_P_STRIDE]

    for (int idx = threadIdx.x; idx < N_PAIRS * OUT_DIM; idx += blockDim.x) {
        const int p = idx >> 7;              // / OUT_DIM
        const int c = idx & (OUT_DIM - 1);
        float2 v;
        v.x = W[(2 * p)     * OUT_DIM + c];
        v.y = W[(2 * p + 1) * OUT_DIM + c];
        ldsP[p * LDS_P_STRIDE + c] = v;
    }
    __syncthreads();

    const int lane  = threadIdx.x & 31;
    const int wave  = threadIdx.x >> 5;
    const int waveN = wave & 1;              // column half: 0 or 1
    const int waveM = wave >> 1;             // 32-row sub-block: 0..3
    const int cBase = waveN * 64;            // in float columns
    const int nBlk  = (nRows + 127) >> 7;    // 128 rows per block iteration

    for (int blk = blockIdx.x; blk < nBlk; blk += gridDim.x) {
        const int mBase = (blk << 7) + waveM * 32;
        if (mBase >= nRows) continue;        // wave-uniform skip
        if (mBase + 32 <= nRows)
            compute_block<false>(X, ldsP, H, mBase, cBase, nRows, lane);
        else
            compute_block<true>(X, ldsP, H, mBase, cBase, nRows, lane);
    }
}

// -------------------------------------------------------------------------
// Kernel 2: out[n, :] = b[:]  (bias epilogue hoisted to init, float4 stores)
// -------------------------------------------------------------------------
__global__ void bias_init(const float* __restrict__ b, float* __restrict__ out,
                          int nRows) {
    const int idx = blockIdx.x * blockDim.x + threadIdx.x;   // one float4 each
    const int total = nRows * (OUT_DIM / 4);
    if (idx < total) {
        const int jg = idx & ((OUT_DIM / 4) - 1);
        const float4 v = *(const float4*)(b + (jg << 2));
        *(float4*)(out + ((size_t)idx << 2)) = v;
    }
}

// -------------------------------------------------------------------------
// Kernel 3: out[dst] += val * h[src]. One wave per edge: lane handles 4
// consecutive floats (full wave = one contiguous 512B row of h, L2-resident).
// unsafeAtomicAdd -> non-returning global_atomic_add_f32 at L2.
// -------------------------------------------------------------------------
__global__ void edge_scatter(const float* __restrict__ H,
                             const int* __restrict__ esrc,
                             const int* __restrict__ edst,
                             const float* __restrict__ evals,
                             float* __restrict__ out, int nEdges) {
    const int lane = threadIdx.x & 31;
    const int wave = threadIdx.x >> 5;
    const int wavesPerBlock = blockDim.x >> 5;
    const int stride = gridDim.x * wavesPerBlock;

    for (int e = blockIdx.x * wavesPerBlock + wave; e < nEdges; e += stride) {
        const int   s = esrc[e];
        const int   d = edst[e];
        const float w = evals[e];
        const float4 hv = *(const float4*)(H + (size_t)s * OUT_DIM + lane * 4);
        float* o = out + (size_t)d * OUT_DIM + lane * 4;
        unsafeAtomicAdd(o + 0, w * hv.x);
        unsafeAtomicAdd(o + 1, w * hv.y);
        unsafeAtomicAdd(o + 2, w * hv.z);
        unsafeAtomicAdd(o + 3, w * hv.w);
    }
}

// -------------------------------------------------------------------------
extern "C" void kernel_launch(void* const* d_in, const int* in_sizes, int n_in,
                              void* d_out, int out_size, void* d_ws, size_t ws_size,
                              hipStream_t stream) {
    const float* feat  = (const float*)d_in[0];   // [N, 256]
    const int*   esrc  = (const int*)  d_in[1];   // [E]
    const int*   edst  = (const int*)  d_in[2];   // [E]
    const float* evals = (const float*)d_in[3];   // [E]
    const float* W     = (const float*)d_in[4];   // [256, 128]
    const float* bias  = (const float*)d_in[5];   // [128]
    float* out = (float*)d_out;                   // [N, 128]
    float* H   = (float*)d_ws;                    // scratch: [N, 128] f32

    const int nRows  = in_sizes[0] / IN_DIM;
    const int nEdges = in_sizes[1];

    // --- dense transform (WMMA f32) ---
    int gemmBlocks = (nRows + 127) / 128;         // one 128x128 super-tile each
    if (gemmBlocks < 1) gemmBlocks = 1;
    const size_t ldsBytes = (size_t)N_PAIRS * LDS_P_STRIDE * sizeof(float2); // 144 KB
    gemm_wmma_f32<<<gemmBlocks, 256, ldsBytes, stream>>>(feat, W, H, nRows);

    // --- bias init of output ---
    const int total4 = nRows * (OUT_DIM / 4);
    bias_init<<<(total4 + 255) / 256, 256, 0, stream>>>(bias, out, nRows);

    // --- edge scatter with f32 atomics ---
    const int wavesPerBlock = 8;                  // 256 threads
    int sBlocks = (nEdges + wavesPerBlock - 1) / wavesPerBlock;
    if (sBlocks < 1) sBlocks = 1;
    edge_scatter<<<sBlocks, 256, 0, stream>>>(H, esrc, edst, evals, out, nEdges);
}